// OriginalEmbedding_14791867368150
// MI455X (gfx1250) — compile-verified
//
#include <hip/hip_runtime.h>
#include <stdint.h>

// Problem constants (from the reference)
#define BATCH 4096
#define SLOTS 26
#define NNZ   10
#define NROWS (BATCH * SLOTS)   // 106496
#define DIM   64                // floats per embedding row (256 bytes)

#define WAVES_PER_BLOCK 8
#define BLOCK_THREADS   (WAVES_PER_BLOCK * 32)   // wave32

// 64-bit async payload type: the builtin is typed as v2i (int vector_size 8).
typedef int v2i __attribute__((ext_vector_type(2)));
typedef __attribute__((address_space(1))) v2i* as1_v2i_ptr;   // global
typedef __attribute__((address_space(3))) v2i* as3_v2i_ptr;   // LDS

#if defined(__has_builtin)
#if __has_builtin(__builtin_amdgcn_global_load_async_to_lds_b64)
#define HAVE_ASYNC_LDS_BUILTIN 1
#endif
#if __has_builtin(__builtin_amdgcn_s_wait_asynccnt)
#define HAVE_WAIT_ASYNC_BUILTIN 1
#endif
#endif

__device__ __forceinline__ void async_gather_b64(const float* gsrc, float* ldst) {
#ifdef HAVE_ASYNC_LDS_BUILTIN
  // Signature (probe-confirmed): (v2i* global src, v2i* lds dst, imm off, imm cpol)
  __builtin_amdgcn_global_load_async_to_lds_b64(
      (as1_v2i_ptr)(uintptr_t)gsrc,
      (as3_v2i_ptr)(uint32_t)(uintptr_t)ldst,
      0, 0);
#else
  uint32_t laddr = (uint32_t)(uintptr_t)ldst;
  uint64_t gaddr = (uint64_t)(uintptr_t)gsrc;
  asm volatile("global_load_async_to_lds_b64 %0, %1, off"
               :: "v"(laddr), "v"(gaddr)
               : "memory");
#endif
}

__device__ __forceinline__ void wait_async_zero() {
#ifdef HAVE_WAIT_ASYNC_BUILTIN
  __builtin_amdgcn_s_wait_asynccnt(0);
#else
  asm volatile("s_wait_asynccnt 0" ::: "memory");
#endif
}

// ---------------------------------------------------------------------------
// One wave (32 lanes) per output row:
//   - row id pinned to an SGPR via readfirstlane -> keys/mask fetched once per
//     wave through SMEM (constant cache), mask tests are scalar branches
//   - each valid nnz: ONE async instruction moves the whole 256B table row
//     into this wave's LDS staging area (32 lanes x 8B), ASYNCcnt-tracked;
//     scalar branch skips masked rows entirely (~30% of HBM gather traffic)
//   - s_wait_asynccnt 0, then each lane reduces its float2 across valid nnz
//     from LDS (a 256B row spans all 64 banks exactly once -> conflict free)
//   - coalesced 256B float2 store of the pooled row
// ---------------------------------------------------------------------------
__global__ __launch_bounds__(BLOCK_THREADS)
void emb_pool_async_kernel(const int* __restrict__ keys,
                           const unsigned char* __restrict__ mask,
                           const float* __restrict__ table,
                           float* __restrict__ out) {
  __shared__ float lbuf[WAVES_PER_BLOCK * NNZ * DIM];   // 8*10*64*4 = 20 KB

  const int lane = (int)(threadIdx.x & 31u);
  const int wave = (int)(threadIdx.x >> 5u);
  // Wave-uniform row index, pinned to an SGPR so the compiler proves
  // uniformity (tid>>5 alone defeats divergence analysis).
  const int n = __builtin_amdgcn_readfirstlane((int)blockIdx.x * WAVES_PER_BLOCK + wave);
  if (n >= NROWS) return;

  const int* krow = keys + (size_t)n * NNZ;
  const unsigned char* mrow = mask + (size_t)n * NNZ;

  // Wave-uniform key values + validity bitmask (SGPRs, SMEM loads).
  int kv[NNZ];
  uint32_t mb = 0;
#pragma unroll
  for (int j = 0; j < NNZ; ++j) {
    kv[j] = krow[j];
    if (mrow[j] != 0) mb |= (1u << j);
  }

  float* wbuf = lbuf + wave * (NNZ * DIM);

  // Issue one async 256B row-gather per *valid* nnz. Lane l moves
  // bytes [8l, 8l+8) of the row. Condition is scalar -> s_cbranch.
#pragma unroll
  for (int j = 0; j < NNZ; ++j) {
    if (mb & (1u << j)) {
      const float* gsrc = table + (size_t)kv[j] * DIM + (size_t)(lane * 2);
      float* ldst = wbuf + j * DIM + lane * 2;
      async_gather_b64(gsrc, ldst);
    }
  }

  // Drain the async engine, then reduce from LDS.
  wait_async_zero();

  float ax = 0.0f, ay = 0.0f;
#pragma unroll
  for (int j = 0; j < NNZ; ++j) {
    if (mb & (1u << j)) {
      ax += wbuf[j * DIM + lane * 2 + 0];
      ay += wbuf[j * DIM + lane * 2 + 1];
    }
  }

  float2 r;
  r.x = ax;
  r.y = ay;
  *(float2*)(out + (size_t)n * DIM + (size_t)(lane * 2)) = r;   // 256B/row store
}

// ---------------------------------------------------------------------------
// Host launcher. Inputs (setup_inputs order): keys int32 [N,NNZ],
// mask bool(1B) [N,NNZ], table f32 [VOCAB,DIM]. Output f32 [N*DIM].
// ---------------------------------------------------------------------------
extern "C" void kernel_launch(void* const* d_in, const int* in_sizes, int n_in,
                              void* d_out, int out_size, void* d_ws, size_t ws_size,
                              hipStream_t stream) {
  (void)in_sizes; (void)n_in; (void)out_size; (void)d_ws; (void)ws_size;
  const int* keys = (const int*)d_in[0];
  const unsigned char* mask = (const unsigned char*)d_in[1];
  const float* table = (const float*)d_in[2];
  float* out = (float*)d_out;

  const int blocks = (NROWS + WAVES_PER_BLOCK - 1) / WAVES_PER_BLOCK;  // 13312
  emb_pool_async_kernel<<<blocks, BLOCK_THREADS, 0, stream>>>(keys, mask, table, out);
}